// JPEG_layer_77996606095536
// MI455X (gfx1250) — compile-verified
//
#include <hip/hip_runtime.h>
#include <hip/hip_bf16.h>

typedef float v2f __attribute__((ext_vector_type(2)));
typedef float v4f __attribute__((ext_vector_type(4)));
typedef float v8f __attribute__((ext_vector_type(8)));

#define IMG_HH 224
#define IMG_WW 224
#define PLANE  (IMG_HH * IMG_WW)
#define TILES_X (IMG_WW / 16)
#define TILES_PER_IMG ((IMG_HH / 16) * (IMG_WW / 16))

// H[a][b] = cos((2a+1)*b*pi/16)  (8x8 DCT-II basis, hardcoded)
__constant__ float HTAB[64] = {
  1.0f,  0.98078528f,  0.92387953f,  0.83146961f,  0.70710678f,  0.55557023f,  0.38268343f,  0.19509032f,
  1.0f,  0.83146961f,  0.38268343f, -0.19509032f, -0.70710678f, -0.98078528f, -0.92387953f, -0.55557023f,
  1.0f,  0.55557023f, -0.38268343f, -0.98078528f, -0.70710678f,  0.19509032f,  0.92387953f,  0.83146961f,
  1.0f,  0.19509032f, -0.92387953f, -0.55557023f,  0.70710678f,  0.83146961f, -0.38268343f, -0.98078528f,
  1.0f, -0.19509032f, -0.92387953f,  0.55557023f,  0.70710678f, -0.83146961f, -0.38268343f,  0.98078528f,
  1.0f, -0.55557023f, -0.38268343f,  0.98078528f, -0.70710678f, -0.19509032f,  0.92387953f, -0.83146961f,
  1.0f, -0.83146961f,  0.38268343f,  0.19509032f, -0.70710678f,  0.98078528f, -0.92387953f,  0.55557023f,
  1.0f, -0.98078528f,  0.92387953f, -0.83146961f,  0.70710678f, -0.55557023f,  0.38268343f, -0.19509032f,
};

// 16x16x16 matmul as 4x V_WMMA_F32_16X16X4_F32.
// A 16x4 frag: lane<16 holds K=kb,kb+1; lane>=16 holds K=kb+2,kb+3 (row m = lane&15).
// B 4x16 frag: symmetric (col n = lane&15). C/D: VGPR v -> row v+8*(lane>>4), col lane&15.
__device__ __forceinline__ v8f mm_constA(const v2f* cA, const float* Bm, int l16, int hi) {
  v2f b[4];
#pragma unroll
  for (int i = 0; i < 4; ++i)
    b[i] = *(const v2f*)(Bm + l16 * 16 + 4 * i + 2 * hi);  // col-major LDS: contiguous
  v8f acc = {};
#pragma unroll
  for (int i = 0; i < 4; ++i)
    acc = __builtin_amdgcn_wmma_f32_16x16x4_f32(false, cA[i], false, b[i], (short)0, acc, false, false);
  return acc;
}
__device__ __forceinline__ v8f mm_constB(const v2f* cB, const float* Am, int l16, int hi) {
  v2f a[4];
#pragma unroll
  for (int i = 0; i < 4; ++i) {
    int k0 = 4 * i + 2 * hi;
    a[i].x = Am[k0 * 16 + l16];
    a[i].y = Am[(k0 + 1) * 16 + l16];
  }
  v8f acc = {};
#pragma unroll
  for (int i = 0; i < 4; ++i)
    acc = __builtin_amdgcn_wmma_f32_16x16x4_f32(false, a[i], false, cB[i], (short)0, acc, false, false);
  return acc;
}

__device__ __forceinline__ void st8(float* p, v8f x) {
  v4f lo = {x[0], x[1], x[2], x[3]};
  v4f hi = {x[4], x[5], x[6], x[7]};
  *(v4f*)p = lo;
  *(v4f*)(p + 4) = hi;
}

// 5-candidate softmax quantizer (NUM_Q=5, Q_HALF=2, clip [-127,123]).
// Hardware v_rcp_f32 instead of IEEE division: grid selection is smoothed by the
// softmax, and the softmax denominator is >= 1 after max-subtraction.
__device__ __forceinline__ float soft_quant(float inp, float qv, float av) {
  float t = rintf(inp * __builtin_amdgcn_rcpf(qv));
  t = fminf(fmaxf(t - 2.0f, -127.0f), 123.0f);
  float iq[5], e[5];
  float m = -3.4e38f;
#pragma unroll
  for (int j = 0; j < 5; ++j) {
    iq[j] = (t + (float)j) * qv;
    float d = iq[j] - inp;
    e[j] = -av * d * d;
    m = fmaxf(m, e[j]);
  }
  float num = 0.0f, den = 0.0f;
#pragma unroll
  for (int j = 0; j < 5; ++j) {
    float w = __expf(e[j] - m);
    den += w;
    num += w * iq[j];
  }
  return num * __builtin_amdgcn_rcpf(den);
}

__global__ __launch_bounds__(256) void jpeg_fused(
    const float* __restrict__ in, const float* __restrict__ qlum,
    const float* __restrict__ qchr, const float* __restrict__ alum,
    const float* __restrict__ achr, float* __restrict__ out, int ntiles) {
  __shared__ float lds[8][4][256];  // per-wave: Y, Cb, Cr tiles (col-major) + scratch
  const int lane = threadIdx.x & 31;
  const int wave = threadIdx.x >> 5;
  const int hi = lane >> 4;
  const int l16 = lane & 15;
  const int jj = l16 & 7;

  const int tile = blockIdx.x * 8 + wave;
  if (tile >= ntiles) return;  // wave-uniform
  const int img = tile / TILES_PER_IMG;
  const int t = tile % TILES_PER_IMG;
  const int row0 = (t / TILES_X) * 16;
  const int col0 = (t % TILES_X) * 16;

  const float* Rp = in + (size_t)(img * 3 + 0) * PLANE;
  const float* Gp = in + (size_t)(img * 3 + 1) * PLANE;
  const float* Bp = in + (size_t)(img * 3 + 2) * PLANE;

  float* bufY = lds[wave][0];
  float* bufCb = lds[wave][1];
  float* bufCr = lds[wave][2];
  float* S = lds[wave][3];

  // Phase A: coalesced float4 tile loads, (x-128) RGB -> YCbCr, scatter col-major to LDS
#pragma unroll
  for (int qq = 0; qq < 2; ++qq) {
    int q = lane + 32 * qq;
    int r = q >> 2, c4 = (q & 3) << 2;
    size_t off = (size_t)(row0 + r) * IMG_WW + col0 + c4;
    v4f r4 = *(const v4f*)(Rp + off);
    v4f g4 = *(const v4f*)(Gp + off);
    v4f b4 = *(const v4f*)(Bp + off);
#pragma unroll
    for (int j = 0; j < 4; ++j) {
      float rr = r4[j] - 128.0f, gg = g4[j] - 128.0f, bb = b4[j] - 128.0f;
      float y = 0.299f * rr + 0.587f * gg + 0.114f * bb;
      float cb = (bb - y) * 0.56433408f + 0.5f;  // 1/(2*(1-0.114))
      float cr = (rr - y) * 0.71326676f + 0.5f;  // 1/(2*(1-0.299))
      int col = c4 + j;
      bufY[col * 16 + r] = y;
      bufCb[col * 16 + r] = cb;
      bufCr[col * 16 + r] = cr;
    }
  }

  // Per-lane constant WMMA fragments for blockdiag(H,H), built once from HTAB.
  // Forward (DCT) frags: element G(k, l16); Reverse (IDCT) frags: element G(l16, k).
  // Same-half mask is identical for k0 and k0+1 (k0 even), applied branchlessly.
  v2f cF[4], cR[4];
#pragma unroll
  for (int i = 0; i < 4; ++i) {
    int k0 = 4 * i + 2 * hi;
    float msk = (((k0 ^ l16) & 8) == 0) ? 1.0f : 0.0f;
    cF[i].x = HTAB[(k0 & 7) * 8 + jj] * msk;
    cF[i].y = HTAB[((k0 + 1) & 7) * 8 + jj] * msk;
    cR[i].x = HTAB[jj * 8 + (k0 & 7)] * msk;
    cR[i].y = HTAB[jj * 8 + ((k0 + 1) & 7)] * msk;
  }

  // Per-lane q/alpha column preload (8x8 tables, lane needs column jj only)
  float ql[8], al[8], qc[8], ac[8];
#pragma unroll
  for (int v = 0; v < 8; ++v) {
    ql[v] = qlum[v * 8 + jj];
    al[v] = alum[v * 8 + jj];
    qc[v] = qchr[v * 8 + jj];
    ac[v] = achr[v * 8 + jj];
  }
  const float nj = (jj == 0) ? 0.70710678f : 1.0f;

  asm volatile("s_wait_dscnt 0" ::: "memory");

#pragma unroll
  for (int c = 0; c < 3; ++c) {
    float* X = lds[wave][c];
    // DCT left:  T = diag(H^T,H^T) @ X   (A[m][k] = G(k,m) -> cF)
    v8f acc = mm_constA(cF, X, l16, hi);
    st8(S + l16 * 16 + 8 * hi, acc);
    // DCT right: C = T @ diag(H,H)       (B[k][n] = G(k,n) -> cF)
    v8f acc2 = mm_constB(cF, S, l16, hi);
    // scale * N, soft-quant, re-apply N for IDCT input
    float yv[8];
#pragma unroll
    for (int v = 0; v < 8; ++v) {
      float nv = ((v == 0) ? 0.70710678f : 1.0f) * nj;
      float inp = 0.25f * nv * acc2[v];
      float qv = (c == 0) ? ql[v] : qc[v];
      float av = (c == 0) ? al[v] : ac[v];
      float rec = soft_quant(inp, qv, av);
      yv[v] = nv * rec;
    }
    {
      v4f lo = {yv[0], yv[1], yv[2], yv[3]};
      v4f hi4 = {yv[4], yv[5], yv[6], yv[7]};
      *(v4f*)(S + l16 * 16 + 8 * hi) = lo;
      *(v4f*)(S + l16 * 16 + 8 * hi + 4) = hi4;
    }
    // IDCT left:  U = diag(H,H) @ Y      (A[m][k] = G(m,k) -> cR)
    v8f acc3 = mm_constA(cR, S, l16, hi);
    st8(S + l16 * 16 + 8 * hi, acc3);
    // IDCT right: R = U @ diag(H^T,H^T)  (B[k][n] = G(n,k) -> cR)
    v8f acc4 = mm_constB(cR, S, l16, hi);
    v8f rec8;
#pragma unroll
    for (int v = 0; v < 8; ++v) rec8[v] = 0.25f * acc4[v];
    st8(X + l16 * 16 + 8 * hi, rec8);  // overwrite channel buffer with reconstruction
  }

  asm volatile("s_wait_dscnt 0" ::: "memory");

  // Phase D: YCbCr -> RGB, +128, /255, normalize; coalesced float4 stores
  const float s0 = 1.0f / (255.0f * 0.2675f), s1 = 1.0f / (255.0f * 0.2565f),
              s2 = 1.0f / (255.0f * 0.2761f);
  const float o0 = (128.0f / 255.0f - 0.5071f) / 0.2675f;
  const float o1 = (128.0f / 255.0f - 0.4867f) / 0.2565f;
  const float o2 = (128.0f / 255.0f - 0.4408f) / 0.2761f;
  float* Ro = out + (size_t)(img * 3 + 0) * PLANE;
  float* Go = out + (size_t)(img * 3 + 1) * PLANE;
  float* Bo = out + (size_t)(img * 3 + 2) * PLANE;
#pragma unroll
  for (int qq = 0; qq < 2; ++qq) {
    int q = lane + 32 * qq;
    int r = q >> 2, c4 = (q & 3) << 2;
    size_t off = (size_t)(row0 + r) * IMG_WW + col0 + c4;
    v4f r4, g4, b4;
#pragma unroll
    for (int j = 0; j < 4; ++j) {
      int col = c4 + j;
      float y = bufY[col * 16 + r];
      float cb = bufCb[col * 16 + r] - 0.5f;
      float cr = bufCr[col * 16 + r] - 0.5f;
      float rr = y + 1.402f * cr;
      float gg = y - 0.71413628f * cr - 0.34413627f * cb;
      float bb = y + 1.772f * cb;
      r4[j] = rr * s0 + o0;
      g4[j] = gg * s1 + o1;
      b4[j] = bb * s2 + o2;
    }
    *(v4f*)(Ro + off) = r4;
    *(v4f*)(Go + off) = g4;
    *(v4f*)(Bo + off) = b4;
  }
}

extern "C" void kernel_launch(void* const* d_in, const int* in_sizes, int n_in,
                              void* d_out, int out_size, void* d_ws, size_t ws_size,
                              hipStream_t stream) {
  (void)n_in; (void)out_size; (void)d_ws; (void)ws_size;
  const float* in = (const float*)d_in[0];
  const float* qlum = (const float*)d_in[1];
  const float* qchr = (const float*)d_in[2];
  const float* alum = (const float*)d_in[3];
  const float* achr = (const float*)d_in[4];
  float* out = (float*)d_out;
  int images = in_sizes[0] / (3 * PLANE);
  int ntiles = images * TILES_PER_IMG;
  int blocks = (ntiles + 7) / 8;  // 8 waves (tiles) per 256-thread workgroup
  jpeg_fused<<<blocks, 256, 0, stream>>>(in, qlum, qchr, alum, achr, out, ntiles);
}